// Transducer_25924422599462
// MI455X (gfx1250) — compile-verified
//
#include <hip/hip_runtime.h>
#include <math.h>

// Sizes (fixed by the reference)
#define NB   8      // batch
#define TT   256    // time
#define UU   64     // target len
#define U1   65     // U+1
#define VV   256    // vocab
#define FEA  1024
#define EMBD 512
#define HID  512
#define G4   2048   // 4*HID
#define MPAD 528    // NB*U1 (=520) padded to multiple of 16
#define NEGV (-1e5f)

typedef __attribute__((ext_vector_type(2))) float v2f;
typedef __attribute__((ext_vector_type(8))) float v8f;
typedef __attribute__((ext_vector_type(4))) unsigned int u32x4;
typedef __attribute__((ext_vector_type(4))) int i32x4;
typedef __attribute__((ext_vector_type(8))) int i32x8;

#if defined(__has_builtin)
#  if __has_builtin(__builtin_amdgcn_tensor_load_to_lds)
#    define HAVE_TDM 1
#  endif
#endif
#ifndef HAVE_TDM
#  define HAVE_TDM 0
#endif

__device__ __forceinline__ v8f wmma_f32_4(v2f a, v2f b, v8f c) {
  // V_WMMA_F32_16X16X4_F32 : D(16x16,f32) = A(16x4,f32) x B(4x16,f32) + C
  return __builtin_amdgcn_wmma_f32_16x16x4_f32(false, a, false, b, (short)0, c,
                                               false, false);
}

#if HAVE_TDM
// Best-effort D# (ISA §8.3/8.4): 2D tile, data_size=4B, tile = row_elems x nrows,
// row pitch = row_stride_elems. Groups 2/3 zero (<=2D tensor).
__device__ __forceinline__ void tdm_load_rows_to_lds(
    const float* gsrc, unsigned lds_byte_off,
    unsigned row_elems, unsigned nrows, unsigned row_stride_elems)
{
  unsigned long long ga = (unsigned long long)(__SIZE_TYPE__)gsrc;
  u32x4 g0;
  g0.x = 1u;                                        // count=1 user descriptor
  g0.y = lds_byte_off;                              // lds_addr
  g0.z = (unsigned)(ga & 0xFFFFFFFFu);              // global_addr[31:0]
  g0.w = (unsigned)((ga >> 32) & 0x01FFFFFFu) | (2u << 30); // [56:32] | type=2
  i32x8 g1;
  g1[0] = (int)(2u << 16);                          // data_size=4B, no multicast
  g1[1] = (int)((row_elems & 0xFFFFu) << 16);       // tensor_dim0[15:0]
  g1[2] = (int)((row_elems >> 16) | ((nrows & 0xFFFFu) << 16)); // dim0 hi|dim1 lo
  g1[3] = (int)((row_elems & 0xFFFFu) << 16);       // dim1 hi=0 | tile_dim0
  g1[4] = (int)(nrows & 0xFFFFu);                   // tile_dim1 (tile_dim2=0)
  g1[5] = (int)row_stride_elems;                    // tensor_dim0_stride[31:0]
  g1[6] = 0;                                        // stride hi | dim1_stride lo
  g1[7] = 0;
  i32x4 z4 = {0, 0, 0, 0};
#if __clang_major__ >= 23
  i32x8 z8 = {0, 0, 0, 0, 0, 0, 0, 0};
  __builtin_amdgcn_tensor_load_to_lds(g0, g1, z4, z4, z8, 0);
#else
  __builtin_amdgcn_tensor_load_to_lds(g0, g1, z4, z4, 0);
#endif
}
#endif

// ---------------------------------------------------------------------------
// C[M,N] = A[M,K] @ W[N,K]^T + bias1 + bias2.
// One wave per PAIR of adjacent 16x16 tiles: one A fragment feeds two
// independent WMMA accumulate chains (halves A traffic, 2x ILP).
// ---------------------------------------------------------------------------
__global__ void __launch_bounds__(256) gemm_wmma_bias(
    const float* __restrict__ A, int lda,
    const float* __restrict__ W,        // (N, K) row-major
    const float* __restrict__ bias1,
    const float* __restrict__ bias2,
    float* __restrict__ C, int ldc,
    int K, int Npairs, int totalPairs)
{
  const int wave = threadIdx.x >> 5;
  const int lane = threadIdx.x & 31;
  const int pair = blockIdx.x * 8 + wave;
  if (pair >= totalPairs) return;          // wave-uniform: EXEC stays full
  const int mt = pair / Npairs, np = pair % Npairs;
  const int m0 = mt * 16, n0 = np * 32;
  const int r    = m0 + (lane & 15);
  const int col0 = n0 + (lane & 15);
  const int koff = (lane >> 4) << 1;
  const float* Arow  = A + (size_t)r * lda + koff;
  const float* Wrow0 = W + (size_t)col0 * K + koff;
  const float* Wrow1 = Wrow0 + (size_t)16 * K;
  __builtin_prefetch(Arow, 0, 3);
  __builtin_prefetch(Wrow0, 0, 3);
  __builtin_prefetch(Wrow1, 0, 3);
  v8f acc0 = {0.f,0.f,0.f,0.f,0.f,0.f,0.f,0.f};
  v8f acc1 = {0.f,0.f,0.f,0.f,0.f,0.f,0.f,0.f};
  for (int k0 = 0; k0 < K; k0 += 4) {
    v2f a  = *(const v2f*)(Arow + k0);
    v2f b0 = *(const v2f*)(Wrow0 + k0);
    v2f b1 = *(const v2f*)(Wrow1 + k0);
    acc0 = wmma_f32_4(a, b0, acc0);
    acc1 = wmma_f32_4(a, b1, acc1);
  }
  float bv0 = bias1 ? bias1[col0] : 0.f;
  float bv1 = bias1 ? bias1[col0 + 16] : 0.f;
  if (bias2) { bv0 += bias2[col0]; bv1 += bias2[col0 + 16]; }
  const int rbase = m0 + ((lane >> 4) << 3);
  #pragma unroll
  for (int i = 0; i < 8; ++i) {
    C[(size_t)(rbase + i) * ldc + col0]      = acc0[i] + bv0;
    C[(size_t)(rbase + i) * ldc + col0 + 16] = acc1[i] + bv1;
  }
}

// ---------------------------------------------------------------------------
// Embedding gather: x[row,d] = emb[token(row), d], rows >= 520 zero-padded.
// ---------------------------------------------------------------------------
__global__ void gather_embed(const int* __restrict__ targets,
                             const float* __restrict__ emb,
                             float* __restrict__ x)
{
  int idx = blockIdx.x * blockDim.x + threadIdx.x;   // MPAD*EMBD
  if (idx >= MPAD * EMBD) return;
  int row = idx >> 9, d = idx & 511;
  float v = 0.f;
  if (row < NB * U1) {
    int n = row / U1, u = row % U1;
    int tok = (u == 0) ? 0 : targets[n * UU + (u - 1)];
    v = emb[(size_t)tok * EMBD + d];
  }
  x[idx] = v;
}

__global__ void zero_pads(float* __restrict__ hs0, float* __restrict__ hs1)
{
  int idx = blockIdx.x * blockDim.x + threadIdx.x;   // 8 pad rows * 512
  if (idx < (MPAD - NB * U1) * HID) {
    hs0[(size_t)NB * U1 * HID + idx] = 0.f;
    hs1[(size_t)NB * U1 * HID + idx] = 0.f;
  }
}

// ---------------------------------------------------------------------------
// One-workgroup LSTM layer: 65 dependent steps.
// - recurrent GEMM h(16x512)@Whh^T via WMMA, 4 independent chains per wave
// - TDM streams the step's xg rows (8 rows x 2048, pitch 65*2048) into LDS,
//   overlapped with the GEMM; fenced with s_wait_tensorcnt.
// LDS: h 32K + gates 64K + c 16K + xg stage 64K = 176 KB (CDNA5 WGP: 320 KB).
// ---------------------------------------------------------------------------
#define LDS_H   0
#define LDS_G   (16 * HID)
#define LDS_C   (16 * HID + NB * G4)
#define LDS_XG  (16 * HID + NB * G4 + NB * HID)
#define LDS_TOT (16 * HID + NB * G4 + NB * HID + NB * G4)

__global__ void __launch_bounds__(1024) lstm_wmma(
    const float* __restrict__ xg,    // (MPAD, 2048), row = n*65+u
    const float* __restrict__ Whh,   // (2048, 512)
    float* __restrict__ hs)          // (MPAD, 512) out, row = n*65+u
{
  extern __shared__ float smem[];
  float* h_s  = smem + LDS_H;        // 16*512 (rows 8..15 stay zero)
  float* g_s  = smem + LDS_G;        // 8*2048
  float* c_s  = smem + LDS_C;        // 8*512
#if HAVE_TDM
  float* xg_s = smem + LDS_XG;       // 8*2048 staged by TDM
#endif
  const int tid  = threadIdx.x;
  const int wave = tid >> 5, lane = tid & 31;
  const int arow = lane & 15;
  const int koff = (lane >> 4) << 1;

  for (int i = tid; i < 16 * HID; i += 1024) h_s[i] = 0.f;
  for (int i = tid; i < NB * HID; i += 1024) c_s[i] = 0.f;
  __syncthreads();

  for (int u = 0; u < U1; ++u) {
#if HAVE_TDM
    if (wave == 0)   // one DMA per step; overlaps with the recurrent GEMM
      tdm_load_rows_to_lds(xg + (size_t)u * G4, (unsigned)(LDS_XG * 4),
                           G4, NB, U1 * G4);
#endif
    // g(16x2048) = h(16x512) @ Whh^T ; 128 col-tiles over 32 waves,
    // 4 tiles per wave sharing one LDS A-fragment load per k-step.
    {
      const float* hrow = h_s + arow * HID + koff;
      const int colb = (wave * 4) * 16 + (lane & 15);
      const float* wr0 = Whh + (size_t)(colb)      * HID + koff;
      const float* wr1 = Whh + (size_t)(colb + 16) * HID + koff;
      const float* wr2 = Whh + (size_t)(colb + 32) * HID + koff;
      const float* wr3 = Whh + (size_t)(colb + 48) * HID + koff;
      v8f a0 = {0.f,0.f,0.f,0.f,0.f,0.f,0.f,0.f};
      v8f a1 = a0, a2 = a0, a3 = a0;
      for (int k0 = 0; k0 < HID; k0 += 4) {
        v2f a = *(const v2f*)(hrow + k0);        // ds_load_b64
        v2f b0 = *(const v2f*)(wr0 + k0);
        v2f b1 = *(const v2f*)(wr1 + k0);
        v2f b2 = *(const v2f*)(wr2 + k0);
        v2f b3 = *(const v2f*)(wr3 + k0);
        a0 = wmma_f32_4(a, b0, a0);
        a1 = wmma_f32_4(a, b1, a1);
        a2 = wmma_f32_4(a, b2, a2);
        a3 = wmma_f32_4(a, b3, a3);
      }
      if (lane < 16) {                           // rows 0..7 = real batches
        #pragma unroll
        for (int i = 0; i < 8; ++i) {
          g_s[i * G4 + colb]      = a0[i];
          g_s[i * G4 + colb + 16] = a1[i];
          g_s[i * G4 + colb + 32] = a2[i];
          g_s[i * G4 + colb + 48] = a3[i];
        }
      }
    }
    __syncthreads();
#if HAVE_TDM
    if (wave == 0) __builtin_amdgcn_s_wait_tensorcnt(0);
    __syncthreads();
#endif
    // gates (PyTorch order i,f,g,o) + state update; 8*512 elements
    for (int e = tid; e < NB * HID; e += 1024) {
      const int n = e >> 9, j = e & 511;
#if HAVE_TDM
      const float* xr = xg_s + (size_t)n * G4;
#else
      const float* xr = xg + ((size_t)(n * U1 + u)) * G4;
#endif
      float gi = g_s[n * G4 + j]           + xr[j];
      float gf = g_s[n * G4 + HID + j]     + xr[HID + j];
      float gg = g_s[n * G4 + 2 * HID + j] + xr[2 * HID + j];
      float go = g_s[n * G4 + 3 * HID + j] + xr[3 * HID + j];
      float c = (1.f / (1.f + __expf(-gf))) * c_s[e]
              + (1.f / (1.f + __expf(-gi))) * tanhf(gg);
      float h = (1.f / (1.f + __expf(-go))) * tanhf(c);
      c_s[e] = c;
      h_s[n * HID + j] = h;
      hs[((size_t)(n * U1 + u)) * HID + j] = h;
    }
    __syncthreads();
  }
}

// ---------------------------------------------------------------------------
// Joint + fused log-softmax: per (n,t) block, 65 LSE reductions over V=256.
// Only blank (N,T,65) and emit (N,T,64) are materialized (never the 136 MB
// joint tensor).
// ---------------------------------------------------------------------------
__global__ void __launch_bounds__(256) joint_lse(
    const float* __restrict__ feat,   // (2048, 256) row = n*T+t
    const float* __restrict__ lm,     // (MPAD, 256) row = n*65+u
    const int* __restrict__ targets,  // (8,64)
    const int* __restrict__ tl,       // (8,)
    float* __restrict__ blank,        // (8,256,65)
    float* __restrict__ emit)         // (8,256,64)
{
  __shared__ float fs[VV];
  const int nt = blockIdx.x;           // n*256 + t
  const int n = nt >> 8;
  const int tid = threadIdx.x;
  fs[tid] = feat[(size_t)nt * VV + tid];
  __syncthreads();
  const int wave = tid >> 5, lane = tid & 31;
  const int tln = tl[n];
  for (int u = wave; u < U1; u += 8) {
    const float* lrow = lm + (size_t)(n * U1 + u) * VV;
    float vals[8];
    float m = -1e30f;
    #pragma unroll
    for (int i = 0; i < 8; ++i) {
      float s = fs[lane + i * 32] + lrow[lane + i * 32];
      vals[i] = s;
      m = fmaxf(m, s);
    }
    #pragma unroll
    for (int off = 16; off > 0; off >>= 1)
      m = fmaxf(m, __shfl_xor(m, off, 32));
    float sum = 0.f;
    #pragma unroll
    for (int i = 0; i < 8; ++i) sum += __expf(vals[i] - m);
    #pragma unroll
    for (int off = 16; off > 0; off >>= 1)
      sum += __shfl_xor(sum, off, 32);
    const float lse = m + __logf(sum);
    if (lane == 0) {
      blank[(size_t)nt * U1 + u] = fs[0] + lrow[0] - lse;
      if (u < UU) {
        const int tg = targets[n * UU + u];
        float e = (u < tln) ? (fs[tg] + lrow[tg] - lse) : NEGV;
        emit[(size_t)nt * UU + u] = e;
      }
    }
  }
}

// ---------------------------------------------------------------------------
// Alpha recursion (reference's cumlogsumexp-row formulation) + mean loss.
// ---------------------------------------------------------------------------
__global__ void alpha_loss(
    const float* __restrict__ blank, const float* __restrict__ emit,
    const int* __restrict__ il, const int* __restrict__ tl,
    float* __restrict__ out)
{
  __shared__ float fin[NB];
  const int n = threadIdx.x;
  if (n < NB) {
    float alpha[U1];
    float fv = 0.f;
    const int iln = il[n], tln = tl[n];
    for (int t = 0; t < TT; ++t) {
      const float* bl_prev = blank + (size_t)(n * TT + (t > 0 ? t - 1 : 0)) * U1;
      const float* em      = emit  + (size_t)(n * TT + t) * UU;
      float E = 0.f, run = NEGV;
      for (int u = 0; u < U1; ++u) {
        const float b_u = (t == 0) ? ((u == 0) ? 0.f : NEGV)
                                   : alpha[u] + bl_prev[u];
        const float x  = b_u - E;
        const float mx = fmaxf(run, x), mn = fminf(run, x);
        run = mx + log1pf(__expf(mn - mx));
        alpha[u] = E + run;
        if (u < UU) E += em[u];
      }
      if (t == iln - 1)
        fv = alpha[tln] + blank[(size_t)(n * TT + t) * U1 + tln];
    }
    fin[n] = fv;
  }
  __syncthreads();
  if (threadIdx.x == 0) {
    float s = 0.f;
    for (int i = 0; i < NB; ++i) s += fin[i];
    out[0] = -s / (float)NB;
  }
}

// ---------------------------------------------------------------------------
extern "C" void kernel_launch(void* const* d_in, const int* in_sizes, int n_in,
                              void* d_out, int out_size, void* d_ws, size_t ws_size,
                              hipStream_t stream)
{
  const float* features = (const float*)d_in[0];
  const int*   targets  = (const int*)  d_in[1];
  const int*   il       = (const int*)  d_in[2];
  const int*   tl       = (const int*)  d_in[3];
  const float* Wc       = (const float*)d_in[4];
  const float* bc       = (const float*)d_in[5];
  const float* emb      = (const float*)d_in[6];
  const float* Wih0     = (const float*)d_in[7];
  const float* Whh0     = (const float*)d_in[8];
  const float* bih0     = (const float*)d_in[9];
  const float* bhh0     = (const float*)d_in[10];
  const float* Wih1     = (const float*)d_in[11];
  const float* Whh1     = (const float*)d_in[12];
  const float* bih1     = (const float*)d_in[13];
  const float* bhh1     = (const float*)d_in[14];
  const float* Wout     = (const float*)d_in[15];
  const float* bout     = (const float*)d_in[16];

  char* p = (char*)d_ws;
  float* x    = (float*)p; p += (size_t)MPAD * EMBD * 4;
  float* xg0  = (float*)p; p += (size_t)MPAD * G4   * 4;
  float* hs0  = (float*)p; p += (size_t)MPAD * HID  * 4;
  float* xg1  = (float*)p; p += (size_t)MPAD * G4   * 4;
  float* hs1  = (float*)p; p += (size_t)MPAD * HID  * 4;
  float* lm   = (float*)p; p += (size_t)MPAD * VV   * 4;
  float* feat = (float*)p; p += (size_t)NB * TT * VV * 4;
  float* blnk = (float*)p; p += (size_t)NB * TT * U1 * 4;
  float* emit = (float*)p; p += (size_t)NB * TT * UU * 4;

  const size_t lstm_lds = (size_t)LDS_TOT * sizeof(float);

  gather_embed<<<(MPAD * EMBD + 255) / 256, 256, 0, stream>>>(targets, emb, x);
  zero_pads<<<16, 256, 0, stream>>>(hs0, hs1);

  { // xg0 = x @ Wih0^T + bih0 + bhh0   (528 x 512 -> 2048)
    int Npairs = G4 / 32, total = (MPAD / 16) * Npairs;   // 33*64 = 2112
    gemm_wmma_bias<<<total / 8, 256, 0, stream>>>(x, EMBD, Wih0, bih0, bhh0,
                                                  xg0, G4, EMBD, Npairs, total);
  }
  lstm_wmma<<<1, 1024, lstm_lds, stream>>>(xg0, Whh0, hs0);
  { // xg1 = hs0 @ Wih1^T + bih1 + bhh1
    int Npairs = G4 / 32, total = (MPAD / 16) * Npairs;
    gemm_wmma_bias<<<total / 8, 256, 0, stream>>>(hs0, HID, Wih1, bih1, bhh1,
                                                  xg1, G4, HID, Npairs, total);
  }
  lstm_wmma<<<1, 1024, lstm_lds, stream>>>(xg1, Whh1, hs1);
  { // lm = hs1 @ Wout^T + bout   (528 x 512 -> 256)
    int Npairs = VV / 32, total = (MPAD / 16) * Npairs;   // 33*8 = 264
    gemm_wmma_bias<<<total / 8, 256, 0, stream>>>(hs1, HID, Wout, bout, nullptr,
                                                  lm, VV, HID, Npairs, total);
  }
  { // feat = features @ Wc^T + bc   (2048 x 1024 -> 256)
    int Npairs = VV / 32, total = (NB * TT / 16) * Npairs; // 128*8 = 1024
    gemm_wmma_bias<<<total / 8, 256, 0, stream>>>(features, FEA, Wc, bc, nullptr,
                                                  feat, VV, FEA, Npairs, total);
  }
  joint_lse<<<NB * TT, 256, 0, stream>>>(feat, lm, targets, tl, blnk, emit);
  alpha_loss<<<1, 64, 0, stream>>>(blnk, emit, il, tl, (float*)d_out);
}